// Point_enc_32650341384578
// MI455X (gfx1250) — compile-verified
//
#include <hip/hip_runtime.h>
#include <hip/hip_bf16.h>
#include <math.h>

typedef float v2f __attribute__((ext_vector_type(2)));
typedef float v8f __attribute__((ext_vector_type(8)));

#define WAVES_PER_BLOCK 4

// --------------------------------------------------------------------------
// layout transposes
// --------------------------------------------------------------------------
__global__ void transpose_cf_to_cl(const float* __restrict__ in, float* __restrict__ out,
                                   int B, int C, int N) {
  int idx = blockIdx.x * blockDim.x + threadIdx.x;
  int total = B * C * N;
  if (idx >= total) return;
  int n = idx % N;
  int c = (idx / N) % C;
  int b = idx / (N * C);
  out[((size_t)b * N + n) * C + c] = in[idx];
}

__global__ void transpose_cl_to_cf(const float* __restrict__ in, float* __restrict__ out,
                                   int B, int S, int C) {
  int idx = blockIdx.x * blockDim.x + threadIdx.x;
  int total = B * S * C;
  if (idx >= total) return;
  int c = idx % C;
  int s = (idx / C) % S;
  int b = idx / (C * S);
  out[((size_t)b * C + c) * S + s] = in[idx];
}

// --------------------------------------------------------------------------
// inverse density: invd[b,i] = 2.5*h*N / sum_j exp(-d2/(2h^2))
// --------------------------------------------------------------------------
__global__ void density_kernel(const float* __restrict__ xyz, float* __restrict__ invd,
                               int B, int N, float h) {
  int b = blockIdx.y;
  int i = blockIdx.x * 256 + threadIdx.x;
  __shared__ float tx[256], ty[256], tz[256];
  float px = 0.f, py = 0.f, pz = 0.f;
  if (i < N) {
    px = xyz[((size_t)b * N + i) * 3 + 0];
    py = xyz[((size_t)b * N + i) * 3 + 1];
    pz = xyz[((size_t)b * N + i) * 3 + 2];
  }
  const float inv2h2 = 1.0f / (2.0f * h * h);
  float s = 0.f;
  for (int j0 = 0; j0 < N; j0 += 256) {
    int j = j0 + threadIdx.x;
    if (j < N) {
      tx[threadIdx.x] = xyz[((size_t)b * N + j) * 3 + 0];
      ty[threadIdx.x] = xyz[((size_t)b * N + j) * 3 + 1];
      tz[threadIdx.x] = xyz[((size_t)b * N + j) * 3 + 2];
    }
    __syncthreads();
    int cnt = (N - j0) < 256 ? (N - j0) : 256;
    for (int t = 0; t < cnt; ++t) {
      float dx = px - tx[t], dy = py - ty[t], dz = pz - tz[t];
      float d2 = dx * dx + dy * dy + dz * dz;
      s += expf(-d2 * inv2h2);
    }
    __syncthreads();
  }
  if (i < N) invd[(size_t)b * N + i] = (2.5f * h * (float)N) / s;
}

// --------------------------------------------------------------------------
// farthest point sampling (sequential); one workgroup per batch; dist in LDS
// --------------------------------------------------------------------------
__global__ void fps_kernel(const float* __restrict__ xyz, int N, int S, int* __restrict__ out) {
  const int b = blockIdx.x;
  const int tid = threadIdx.x;
  const int bs = blockDim.x;  // 256
  __shared__ float dist[8192];
  __shared__ float rv[256];
  __shared__ int ri[256];
  __shared__ int curFar;
  for (int i = tid; i < N; i += bs) dist[i] = 1e10f;
  if (tid == 0) curFar = 0;
  __syncthreads();
  for (int it = 0; it < S; ++it) {
    int far = curFar;
    if (tid == 0) out[b * S + it] = far;
    float cx = xyz[((size_t)b * N + far) * 3 + 0];
    float cy = xyz[((size_t)b * N + far) * 3 + 1];
    float cz = xyz[((size_t)b * N + far) * 3 + 2];
    float bv = -1e30f;
    int bi = 0x7fffffff;
    for (int i = tid; i < N; i += bs) {
      float dx = xyz[((size_t)b * N + i) * 3 + 0] - cx;
      float dy = xyz[((size_t)b * N + i) * 3 + 1] - cy;
      float dz = xyz[((size_t)b * N + i) * 3 + 2] - cz;
      float d = dx * dx + dy * dy + dz * dz;
      float dv = dist[i];
      if (d < dv) { dv = d; dist[i] = dv; }
      if (dv > bv || (dv == bv && i < bi)) { bv = dv; bi = i; }
    }
    rv[tid] = bv; ri[tid] = bi;
    __syncthreads();
    for (int d = bs >> 1; d > 0; d >>= 1) {
      if (tid < d) {
        float ov = rv[tid + d]; int oi = ri[tid + d];
        if (ov > rv[tid] || (ov == rv[tid] && oi < ri[tid])) { rv[tid] = ov; ri[tid] = oi; }
      }
      __syncthreads();
    }
    if (tid == 0) curFar = ri[0];
    __syncthreads();
  }
}

__global__ void gather_centers_kernel(const float* __restrict__ xyz, const int* __restrict__ idx,
                                      float* __restrict__ nxyz, int B, int N, int S) {
  int t = blockIdx.x * blockDim.x + threadIdx.x;
  if (t >= B * S) return;
  int b = t / S;
  int j = idx[t];
  nxyz[(size_t)t * 3 + 0] = xyz[((size_t)b * N + j) * 3 + 0];
  nxyz[(size_t)t * 3 + 1] = xyz[((size_t)b * N + j) * 3 + 1];
  nxyz[(size_t)t * 3 + 2] = xyz[((size_t)b * N + j) * 3 + 2];
}

// --------------------------------------------------------------------------
// kNN (K smallest distances): one workgroup per query; dist array in LDS,
// K rounds of argmin-with-exclusion (tie -> lowest index, matches top_k)
// --------------------------------------------------------------------------
__global__ void knn_kernel(const float* __restrict__ xyz, const float* __restrict__ nxyz,
                           int N, int S, int Ksel, int* __restrict__ out) {
  int b = blockIdx.y, s = blockIdx.x;
  int tid = threadIdx.x;
  const int bs = 256;
  __shared__ float d2[8192];
  __shared__ float rv[256];
  __shared__ int ri[256];
  float qx = nxyz[((size_t)b * S + s) * 3 + 0];
  float qy = nxyz[((size_t)b * S + s) * 3 + 1];
  float qz = nxyz[((size_t)b * S + s) * 3 + 2];
  for (int i = tid; i < N; i += bs) {
    float dx = xyz[((size_t)b * N + i) * 3 + 0] - qx;
    float dy = xyz[((size_t)b * N + i) * 3 + 1] - qy;
    float dz = xyz[((size_t)b * N + i) * 3 + 2] - qz;
    d2[i] = dx * dx + dy * dy + dz * dz;
  }
  __syncthreads();
  for (int t = 0; t < Ksel; ++t) {
    float bv = 3.0e38f;
    int bi = 0x7fffffff;
    for (int i = tid; i < N; i += bs) {
      float v = d2[i];
      if (v < bv || (v == bv && i < bi)) { bv = v; bi = i; }
    }
    rv[tid] = bv; ri[tid] = bi;
    __syncthreads();
    for (int d = bs >> 1; d > 0; d >>= 1) {
      if (tid < d) {
        float ov = rv[tid + d]; int oi = ri[tid + d];
        if (ov < rv[tid] || (ov == rv[tid] && oi < ri[tid])) { rv[tid] = ov; ri[tid] = oi; }
      }
      __syncthreads();
    }
    if (tid == 0) {
      out[(((size_t)b * S) + s) * Ksel + t] = ri[0];
      d2[ri[0]] = 3.4e38f;
    }
    __syncthreads();
  }
}

// --------------------------------------------------------------------------
// grouping (element-wise, coalesced):
//   X0[r, 0:3] = xyz[j]-new_xyz ; X0[r, 3:3+D] = pts[j]
// --------------------------------------------------------------------------
__global__ void group_feat_kernel(const float* __restrict__ xyz, const float* __restrict__ nxyz,
                                  const float* __restrict__ pts, const int* __restrict__ knn,
                                  float* __restrict__ X0, int B, int Nin, int S, int Ksel, int D) {
  int idx = blockIdx.x * blockDim.x + threadIdx.x;
  int Cin = 3 + D;
  int total = B * S * Ksel * Cin;
  if (idx >= total) return;
  int c = idx % Cin;
  int r = idx / Cin;
  int j = knn[r];
  int s = (r / Ksel) % S;
  int b = r / (Ksel * S);
  float v;
  if (c < 3)
    v = xyz[((size_t)b * Nin + j) * 3 + c] - nxyz[((size_t)b * S + s) * 3 + c];
  else
    v = pts[((size_t)b * Nin + j) * D + (c - 3)];
  X0[idx] = v;
}

__global__ void group_density_kernel(const float* __restrict__ invd, const int* __restrict__ knn,
                                     float* __restrict__ gd, int B, int Nin, int S, int Ksel) {
  int r = blockIdx.x * blockDim.x + threadIdx.x;
  int total = B * S * Ksel;
  if (r >= total) return;
  int b = r / (Ksel * S);
  gd[r] = invd[(size_t)b * Nin + knn[r]];
}

// normalize grouped density by per-(b,s) max over K (in place)
__global__ void dsnorm_kernel(float* __restrict__ gd, int BS, int Ksel) {
  int idx = blockIdx.x * blockDim.x + threadIdx.x;
  if (idx >= BS) return;
  size_t base = (size_t)idx * Ksel;
  float m = -1e30f;
  for (int k = 0; k < Ksel; ++k) m = fmaxf(m, gd[base + k]);
  float inv = 1.0f / m;
  for (int k = 0; k < Ksel; ++k) gd[base + k] *= inv;
}

// x[r,c] *= ds[r]
__global__ void scale_kernel(float* __restrict__ X, const float* __restrict__ ds, int R, int C) {
  int idx = blockIdx.x * blockDim.x + threadIdx.x;
  if (idx >= R * C) return;
  X[idx] *= ds[idx / C];
}

// --------------------------------------------------------------------------
// generic WMMA FP32 GEMM: Y[M,N] = A[M,K] @ B[K,N] (+ bias[N])
// one wave per 16x16 tile; V_WMMA_F32_16X16X4_F32 over K in steps of 4.
// Branchless: M ragged -> clamp row for loads (extra rows never stored);
// K ragged -> unpredicated steady loop + one value-selected tail step.
// N must be a multiple of 16.
// --------------------------------------------------------------------------
__global__ void gemm_wmma_kernel(const float* __restrict__ A, const float* __restrict__ Bm,
                                 const float* __restrict__ bias, float* __restrict__ Y,
                                 int M, int N, int K, int lda, int ldb, int ldy) {
  int lane = threadIdx.x & 31;
  int wave = threadIdx.x >> 5;
  int nMt = (M + 15) >> 4, nNt = N >> 4;
  int tile = blockIdx.x * WAVES_PER_BLOCK + wave;
  if (tile >= nMt * nNt) return;
  int tm = tile / nNt, tn = tile % nNt;
  int lm = lane & 15;
  int mrow = tm * 16 + lm;
  if (mrow > M - 1) mrow = M - 1;  // clamp: those lanes feed rows that are never stored
  int n = tn * 16 + lm;
  int kg = (lane >> 4) << 1;  // 0 or 2
  const float* Arow = A + (size_t)mrow * lda;
  const float* Bcol = Bm + n;
  v8f acc = {0.f, 0.f, 0.f, 0.f, 0.f, 0.f, 0.f, 0.f};
  int kfull = K & ~3;
  if ((lda & 1) == 0) {  // uniform branch: A pair is 8B-aligned -> b64 load
    for (int kb = 0; kb < kfull; kb += 4) {
      int k0 = kb + kg;
      v2f a = *(const v2f*)(Arow + k0);
      v2f b;
      b.x = Bcol[(size_t)k0 * ldb];
      b.y = Bcol[(size_t)(k0 + 1) * ldb];
      acc = __builtin_amdgcn_wmma_f32_16x16x4_f32(false, a, false, b, (short)0, acc, false, false);
    }
  } else {
    for (int kb = 0; kb < kfull; kb += 4) {
      int k0 = kb + kg;
      v2f a, b;
      a.x = Arow[k0];
      a.y = Arow[k0 + 1];
      b.x = Bcol[(size_t)k0 * ldb];
      b.y = Bcol[(size_t)(k0 + 1) * ldb];
      acc = __builtin_amdgcn_wmma_f32_16x16x4_f32(false, a, false, b, (short)0, acc, false, false);
    }
  }
  if (kfull < K) {  // uniform tail: clamped addresses, value selects (no control flow)
    int k0 = kfull + kg, k1 = k0 + 1;
    int k0c = k0 < K ? k0 : K - 1;
    int k1c = k1 < K ? k1 : K - 1;
    v2f a, b;
    a.x = Arow[k0c];
    a.y = Arow[k1c];
    b.x = Bcol[(size_t)k0c * ldb];
    b.y = Bcol[(size_t)k1c * ldb];
    a.x = (k0 < K) ? a.x : 0.f;
    b.x = (k0 < K) ? b.x : 0.f;
    a.y = (k1 < K) ? a.y : 0.f;
    b.y = (k1 < K) ? b.y : 0.f;
    acc = __builtin_amdgcn_wmma_f32_16x16x4_f32(false, a, false, b, (short)0, acc, false, false);
  }
  float bv = bias ? bias[n] : 0.f;
  int smb = tm * 16 + ((lane >> 4) << 3);
#pragma unroll
  for (int r = 0; r < 8; ++r) {
    int sm = smb + r;
    if (sm < M) Y[(size_t)sm * ldy + n] = acc[r] + bv;
  }
}

// --------------------------------------------------------------------------
// PointConv aggregation: per sampled point, Y[c,w] = sum_k X[k,c]*W[k,w]
// (M=C tile, N=16, K=38) via WMMA f32; one wave per (s, c-tile)
// --------------------------------------------------------------------------
__global__ void pointconv_wmma_kernel(const float* __restrict__ X, const float* __restrict__ Wn,
                                      float* __restrict__ Ypc, int Stot, int C, int K) {
  int lane = threadIdx.x & 31;
  int wave = threadIdx.x >> 5;
  int nCt = C >> 4;
  int tile = blockIdx.x * WAVES_PER_BLOCK + wave;
  if (tile >= Stot * nCt) return;
  int s = tile / nCt, ct = tile % nCt;
  const float* Xs = X + (size_t)s * K * C;
  const float* Ws = Wn + (size_t)s * K * 16;
  int lm = lane & 15;
  int cM = ct * 16 + lm;
  int kg = (lane >> 4) << 1;
  v8f acc = {0.f, 0.f, 0.f, 0.f, 0.f, 0.f, 0.f, 0.f};
  int kfull = K & ~3;
  for (int kb = 0; kb < kfull; kb += 4) {
    int k0 = kb + kg;
    v2f a, b;
    a.x = Xs[(size_t)k0 * C + cM];
    a.y = Xs[(size_t)(k0 + 1) * C + cM];
    b.x = Ws[k0 * 16 + lm];
    b.y = Ws[(k0 + 1) * 16 + lm];
    acc = __builtin_amdgcn_wmma_f32_16x16x4_f32(false, a, false, b, (short)0, acc, false, false);
  }
  if (kfull < K) {
    int k0 = kfull + kg, k1 = k0 + 1;
    int k0c = k0 < K ? k0 : K - 1;
    int k1c = k1 < K ? k1 : K - 1;
    v2f a, b;
    a.x = Xs[(size_t)k0c * C + cM];
    a.y = Xs[(size_t)k1c * C + cM];
    b.x = Ws[k0c * 16 + lm];
    b.y = Ws[k1c * 16 + lm];
    a.x = (k0 < K) ? a.x : 0.f;
    b.x = (k0 < K) ? b.x : 0.f;
    a.y = (k1 < K) ? a.y : 0.f;
    b.y = (k1 < K) ? b.y : 0.f;
    acc = __builtin_amdgcn_wmma_f32_16x16x4_f32(false, a, false, b, (short)0, acc, false, false);
  }
  int cmb = ct * 16 + ((lane >> 4) << 3);
#pragma unroll
  for (int r = 0; r < 8; ++r)
    Ypc[((size_t)s * C + (cmb + r)) * 16 + lm] = acc[r];
}

// --------------------------------------------------------------------------
// small scalar GEMM (WeightNet / DensityNet; N in {1,8,16})
// --------------------------------------------------------------------------
__global__ void gemm_small_kernel(const float* __restrict__ A, const float* __restrict__ W,
                                  const float* __restrict__ bias, float* __restrict__ Y,
                                  int M, int N, int K, int lda, int ldy) {
  int idx = blockIdx.x * blockDim.x + threadIdx.x;
  if (idx >= M * N) return;
  int m = idx / N, n = idx % N;
  float s = bias ? bias[n] : 0.f;
  const float* ar = A + (size_t)m * lda;
  for (int k = 0; k < K; ++k) s += ar[k] * W[k * N + n];
  Y[(size_t)m * ldy + n] = s;
}

// --------------------------------------------------------------------------
// per-channel training-mode BN: stats then normalize+activation
// --------------------------------------------------------------------------
__global__ void col_stats_kernel(const float* __restrict__ Y, int M, int N,
                                 float* __restrict__ mean, float* __restrict__ var) {
  int c = blockIdx.x;
  int tid = threadIdx.x;
  __shared__ float s1[256], s2[256];
  float a = 0.f, q = 0.f;
  for (int i = tid; i < M; i += 256) {
    float v = Y[(size_t)i * N + c];
    a += v; q += v * v;
  }
  s1[tid] = a; s2[tid] = q;
  __syncthreads();
  for (int d = 128; d > 0; d >>= 1) {
    if (tid < d) { s1[tid] += s1[tid + d]; s2[tid] += s2[tid + d]; }
    __syncthreads();
  }
  if (tid == 0) {
    float mu = s1[0] / (float)M;
    mean[c] = mu;
    var[c] = s2[0] / (float)M - mu * mu;
  }
}

__global__ void bn_act_kernel(float* __restrict__ Y, const float* __restrict__ mean,
                              const float* __restrict__ var, const float* __restrict__ g,
                              const float* __restrict__ be, int total, int N, int act) {
  int idx = blockIdx.x * blockDim.x + threadIdx.x;
  if (idx >= total) return;
  int c = idx % N;
  float v = Y[idx];
  v = g[c] * (v - mean[c]) * rsqrtf(var[c] + 1e-5f) + be[c];
  if (act == 1) v = v > 0.f ? v : 0.f;
  else if (act == 2) v = 1.f / (1.f + expf(-v));
  Y[idx] = v;
}

// --------------------------------------------------------------------------
// host orchestration
// --------------------------------------------------------------------------
extern "C" void kernel_launch(void* const* d_in, const int* in_sizes, int n_in,
                              void* d_out, int out_size, void* d_ws, size_t ws_size,
                              hipStream_t stream) {
  (void)in_sizes; (void)n_in; (void)out_size; (void)ws_size;
  const int B = 2, N0 = 8192, K = 38;
  const int NPTS[4] = {1024, 256, 64, 36};
  const float BW[4] = {0.05f, 0.1f, 0.2f, 0.4f};
  const int MLPC[4][3] = {{32, 32, 64}, {64, 64, 128}, {128, 128, 256}, {256, 256, 512}};

  const float* xyz_in = (const float*)d_in[0];  // [B,3,N0]
  float* outp = (float*)d_out;

  float* wsf = (float*)d_ws;
  size_t off = 0;
  auto allocf = [&](size_t n) -> float* { float* p = wsf + off; off += n; return p; };

  // persistent ping-pong buffers across layers
  float* xyz_cl = allocf((size_t)B * N0 * 3);
  float* xyzSlot[2] = {allocf((size_t)B * 1024 * 3), allocf((size_t)B * 1024 * 3)};
  float* ptsSlot[2] = {allocf((size_t)B * 1024 * 64), allocf((size_t)B * 1024 * 64)};
  size_t scratchBase = off;

  {  // [B,3,N] -> [B,N,3]
    int total = B * 3 * N0;
    transpose_cf_to_cl<<<(total + 255) / 256, 256, 0, stream>>>(xyz_in, xyz_cl, B, 3, N0);
  }

  const float* curXyz = xyz_cl;
  const float* curPts = nullptr;
  int Nin = N0, D = 0;
  size_t outOff = 0;
  int pbase = 2;  // d_in index of first param array (after xyz, feat)

  auto gemmW = [&](const float* A, const float* Bm, const float* bias, float* Yo,
                   int M, int Nc, int Kc, int lda, int ldb, int ldy) {
    int tiles = ((M + 15) / 16) * (Nc / 16);
    int blocks = (tiles + WAVES_PER_BLOCK - 1) / WAVES_PER_BLOCK;
    gemm_wmma_kernel<<<blocks, 32 * WAVES_PER_BLOCK, 0, stream>>>(A, Bm, bias, Yo, M, Nc, Kc, lda, ldb, ldy);
  };
  auto gemmS = [&](const float* A, const float* W, const float* bias, float* Yo,
                   int M, int Nc, int Kc, int lda, int ldy) {
    int tot = M * Nc;
    gemm_small_kernel<<<(tot + 255) / 256, 256, 0, stream>>>(A, W, bias, Yo, M, Nc, Kc, lda, ldy);
  };

  for (int L = 0; L < 4; ++L) {
    off = scratchBase;  // reset bump allocator each layer
    int S = NPTS[L];
    int Cin = 3 + D;
    int R = B * S * K;
    int C1 = MLPC[L][0], C2 = MLPC[L][1], C3 = MLPC[L][2];

    // params: per-layer order = mlp(3x{W,b,g,be}), weight(3x4), density(3x4),
    //         linear{W,b}, bn_linear{g,be}
    const float *mW[3], *mB[3], *mG[3], *mBe[3];
    const float *wW[3], *wB[3], *wG[3], *wBe[3];
    const float *dW[3], *dB[3], *dG[3], *dBe[3];
    for (int j = 0; j < 3; ++j) {
      mW[j] = (const float*)d_in[pbase + 4 * j + 0];
      mB[j] = (const float*)d_in[pbase + 4 * j + 1];
      mG[j] = (const float*)d_in[pbase + 4 * j + 2];
      mBe[j] = (const float*)d_in[pbase + 4 * j + 3];
      wW[j] = (const float*)d_in[pbase + 12 + 4 * j + 0];
      wB[j] = (const float*)d_in[pbase + 12 + 4 * j + 1];
      wG[j] = (const float*)d_in[pbase + 12 + 4 * j + 2];
      wBe[j] = (const float*)d_in[pbase + 12 + 4 * j + 3];
      dW[j] = (const float*)d_in[pbase + 24 + 4 * j + 0];
      dB[j] = (const float*)d_in[pbase + 24 + 4 * j + 1];
      dG[j] = (const float*)d_in[pbase + 24 + 4 * j + 2];
      dBe[j] = (const float*)d_in[pbase + 24 + 4 * j + 3];
    }
    const float* lW = (const float*)d_in[pbase + 36];
    const float* lB = (const float*)d_in[pbase + 37];
    const float* lG = (const float*)d_in[pbase + 38];
    const float* lBe = (const float*)d_in[pbase + 39];

    // scratch
    float* invd = allocf((size_t)B * Nin);
    int* fpsIdx = (int*)allocf((size_t)B * S);
    int* knnIdx = (int*)allocf((size_t)R);
    float* newXyz = xyzSlot[L & 1];
    float* X0 = allocf((size_t)R * Cin);
    float* gd = allocf((size_t)R);
    float* Z1 = allocf((size_t)R * C1);
    float* Z2 = allocf((size_t)R * C2);
    float* Z3 = allocf((size_t)R * C3);
    float* ds1 = allocf((size_t)R * 16);
    float* ds2 = allocf((size_t)R * 8);
    float* ds3 = allocf((size_t)R);
    float* w1 = allocf((size_t)R * 8);
    float* w2 = allocf((size_t)R * 8);
    float* w3 = allocf((size_t)R * 16);
    float* ypc = allocf((size_t)B * S * 16 * C3);
    float* mean = allocf(512);
    float* var = allocf(512);
    float* Y = ptsSlot[L & 1];

    auto bn = [&](float* Yo, int M, int Nc, const float* g, const float* be, int act) {
      col_stats_kernel<<<Nc, 256, 0, stream>>>(Yo, M, Nc, mean, var);
      int tot = M * Nc;
      bn_act_kernel<<<(tot + 255) / 256, 256, 0, stream>>>(Yo, mean, var, g, be, tot, Nc, act);
    };

    // 1) inverse density
    density_kernel<<<dim3((Nin + 255) / 256, B), 256, 0, stream>>>(curXyz, invd, B, Nin, BW[L]);
    // 2) farthest point sampling
    fps_kernel<<<B, 256, 0, stream>>>(curXyz, Nin, S, fpsIdx);
    // 3) gather sampled centers
    gather_centers_kernel<<<(B * S + 255) / 256, 256, 0, stream>>>(curXyz, fpsIdx, newXyz, B, Nin, S);
    // 4) kNN
    knn_kernel<<<dim3(S, B), 256, 0, stream>>>(curXyz, newXyz, Nin, S, K, knnIdx);
    // 5) group features + density (element-wise for coalescing)
    {
      int tot = R * Cin;
      group_feat_kernel<<<(tot + 255) / 256, 256, 0, stream>>>(curXyz, newXyz, curPts, knnIdx,
                                                               X0, B, Nin, S, K, D);
      group_density_kernel<<<(R + 255) / 256, 256, 0, stream>>>(invd, knnIdx, gd, B, Nin, S, K);
    }
    // 6) pointwise MLP (GEMM + BN + ReLU) x3  -- WMMA f32
    gemmW(X0, mW[0], mB[0], Z1, R, C1, Cin, Cin, C1, C1); bn(Z1, R, C1, mG[0], mBe[0], 1);
    gemmW(Z1, mW[1], mB[1], Z2, R, C2, C1, C1, C2, C2);   bn(Z2, R, C2, mG[1], mBe[1], 1);
    gemmW(Z2, mW[2], mB[2], Z3, R, C3, C2, C2, C3, C3);   bn(Z3, R, C3, mG[2], mBe[2], 1);
    // 7) density scale net: gd/max_k -> 1->16->8->1 (relu,relu,sigmoid), then x *= ds
    dsnorm_kernel<<<(B * S + 255) / 256, 256, 0, stream>>>(gd, B * S, K);
    gemmS(gd, dW[0], dB[0], ds1, R, 16, 1, 1, 16);  bn(ds1, R, 16, dG[0], dBe[0], 1);
    gemmS(ds1, dW[1], dB[1], ds2, R, 8, 16, 16, 8); bn(ds2, R, 8, dG[1], dBe[1], 1);
    gemmS(ds2, dW[2], dB[2], ds3, R, 1, 8, 8, 1);   bn(ds3, R, 1, dG[2], dBe[2], 2);
    scale_kernel<<<(R * C3 + 255) / 256, 256, 0, stream>>>(Z3, ds3, R, C3);
    // 8) WeightNet on grouped_xyz_norm (= first 3 channels of X0): 3->8->8->16
    gemmS(X0, wW[0], wB[0], w1, R, 8, 3, Cin, 8); bn(w1, R, 8, wG[0], wBe[0], 1);
    gemmS(w1, wW[1], wB[1], w2, R, 8, 8, 8, 8);   bn(w2, R, 8, wG[1], wBe[1], 1);
    gemmS(w2, wW[2], wB[2], w3, R, 16, 8, 8, 16); bn(w3, R, 16, wG[2], wBe[2], 1);
    // 9) PointConv aggregation: y[s, c*16+w] = sum_k x[s,k,c]*w[s,k,w] -- WMMA f32
    {
      int tiles = B * S * (C3 / 16);
      pointconv_wmma_kernel<<<(tiles + WAVES_PER_BLOCK - 1) / WAVES_PER_BLOCK,
                              32 * WAVES_PER_BLOCK, 0, stream>>>(Z3, w3, ypc, B * S, C3, K);
    }
    // 10) linear [BS,16*C3]@[16*C3,C3] + bias, then BN+ReLU -- WMMA f32
    gemmW(ypc, lW, lB, Y, B * S, C3, 16 * C3, 16 * C3, C3, C3);
    bn(Y, B * S, C3, lG, lBe, 1);
    // 11) emit outputs (channels-first)
    {
      int tot = B * S * 3;
      transpose_cl_to_cf<<<(tot + 255) / 256, 256, 0, stream>>>(newXyz, outp + outOff, B, S, 3);
      outOff += (size_t)tot;
    }
    {
      int tot = B * S * C3;
      transpose_cl_to_cf<<<(tot + 255) / 256, 256, 0, stream>>>(Y, outp + outOff, B, S, C3);
      outOff += (size_t)tot;
    }
    // next layer inputs
    curXyz = newXyz;
    curPts = Y;
    Nin = S;
    D = C3;
    pbase += 40;
  }
}